// AdditiveScorer_72206990180539
// MI455X (gfx1250) — compile-verified
//
#include <hip/hip_runtime.h>

typedef __attribute__((ext_vector_type(2))) float v2f;
typedef __attribute__((ext_vector_type(8))) float v8f;

#define D_DIM 512
#define H_DIM 128
#define LQ_DIM 512
#define LK_DIM 512
#define BATCH 4
#define MROWS (BATCH * LQ_DIM) // 2048 projected rows per matrix

// gfx1250 hardware tanh (TRANS pipe, co-executes with VALU)
__device__ __forceinline__ float fast_tanh(float x) {
#if __has_builtin(__builtin_amdgcn_tanhf)
  return __builtin_amdgcn_tanhf(x);
#else
  float y;
  asm volatile("v_tanh_f32 %0, %1" : "=v"(y) : "v"(x));
  return y;
#endif
}

// ---------------------------------------------------------------------------
// Stage 1: P = X @ W via V_WMMA_F32_16X16X4_F32.
// Grid: (MROWS/16, 2) — y=0 projects qs with Wq, y=1 projects ks with Wk.
// Block: 256 threads = 8 waves; wave w owns N-tile [16w, 16w+16) of H=128.
// ---------------------------------------------------------------------------
__global__ __launch_bounds__(256) void proj_wmma_f32(
    const float* __restrict__ qs, const float* __restrict__ ks,
    const float* __restrict__ Wq, const float* __restrict__ Wk,
    float* __restrict__ Qp, float* __restrict__ Kp) {
  const int tid  = threadIdx.x;
  const int lane = tid & 31;
  const int wave = tid >> 5;          // 0..7 -> N tile of 16
  const int m0   = blockIdx.x * 16;
  const int n0   = wave * 16;

  const bool isQ = (blockIdx.y == 0);
  const float* __restrict__ X = isQ ? qs : ks;
  const float* __restrict__ W = isQ ? Wq : Wk;
  float* __restrict__ P       = isQ ? Qp : Kp;

  const int r    = lane & 15;         // row (A) / col (B) within tile
  const int koff = (lane >> 4) << 1;  // lanes 0-15 -> K{0,1}; 16-31 -> K{2,3}

  const float* __restrict__ xrow = X + (size_t)(m0 + r) * D_DIM;

  v8f acc = {};
  for (int k0 = 0; k0 < D_DIM; k0 += 4) {
    v2f a, b;
    a.x = xrow[k0 + koff];
    a.y = xrow[k0 + koff + 1];
    b.x = W[(size_t)(k0 + koff)     * H_DIM + n0 + r];
    b.y = W[(size_t)(k0 + koff + 1) * H_DIM + n0 + r];
    // 8 args: (neg_a, A, neg_b, B, c_mod, C, reuse_a, reuse_b)
    acc = __builtin_amdgcn_wmma_f32_16x16x4_f32(
        false, a, false, b, (short)0, acc, false, false);
  }

  // C/D layout: VGPR i -> M=i (lanes 0-15) / M=i+8 (lanes 16-31), N = lane&15
#pragma unroll
  for (int i = 0; i < 8; ++i) {
    const int row = (lane < 16) ? i : (i + 8);
    P[(size_t)(m0 + row) * H_DIM + n0 + r] = acc[i];
  }
}

// ---------------------------------------------------------------------------
// Stage 2: scores[b,q,k] = sum_h tanh(Qp[b,q,h] + Kp[b,k,h]) * wv[h]
// Block computes a 16(q) x 64(k) tile; Q/K/wv tiles staged in LDS.
// 256 threads: thread (ql = tid&15, kg = tid>>4) computes 4 consecutive k.
// ---------------------------------------------------------------------------
__global__ __launch_bounds__(256) void additive_score(
    const float* __restrict__ Qp, const float* __restrict__ Kp,
    const float* __restrict__ wv, float* __restrict__ out) {
  __shared__ float Qs[16 * H_DIM];   //  8 KB
  __shared__ float Ks[64 * H_DIM];   // 32 KB
  __shared__ float Ws[H_DIM];        // 512 B

  const int tid = threadIdx.x;
  const int b   = blockIdx.z;
  const int q0  = blockIdx.y * 16;
  const int k0  = blockIdx.x * 64;

  const size_t qbase = ((size_t)b * LQ_DIM + q0) * H_DIM;
  const size_t kbase = ((size_t)b * LK_DIM + k0) * H_DIM;

  // Stage tiles into LDS with float4 (b128) transfers.
  {
    const float4* src = (const float4*)(Qp + qbase);
    float4* dst = (float4*)Qs;                     // 512 float4
#pragma unroll
    for (int i = 0; i < 2; ++i) dst[tid + 256 * i] = src[tid + 256 * i];
  }
  {
    const float4* src = (const float4*)(Kp + kbase);
    float4* dst = (float4*)Ks;                     // 2048 float4
#pragma unroll
    for (int i = 0; i < 8; ++i) dst[tid + 256 * i] = src[tid + 256 * i];
  }
  if (tid < 32) ((float4*)Ws)[tid] = ((const float4*)wv)[tid];
  __syncthreads();

  const int ql = tid & 15;
  const int kg = tid >> 4;  // 0..15, each owns 4 k's
  const float* __restrict__ qrow = Qs + ql * H_DIM;

  float acc[4] = {0.f, 0.f, 0.f, 0.f};
  for (int h = 0; h < H_DIM; h += 4) {
    const float4 qv = *(const float4*)(qrow + h);
    const float4 wv4 = *(const float4*)(Ws + h);
#pragma unroll
    for (int j = 0; j < 4; ++j) {
      const float4 kv = *(const float4*)(Ks + (kg * 4 + j) * H_DIM + h);
      acc[j] += fast_tanh(qv.x + kv.x) * wv4.x
              + fast_tanh(qv.y + kv.y) * wv4.y
              + fast_tanh(qv.z + kv.z) * wv4.z
              + fast_tanh(qv.w + kv.w) * wv4.w;
    }
  }

  const size_t obase =
      ((size_t)b * LQ_DIM + (q0 + ql)) * (size_t)LK_DIM + k0 + kg * 4;
#pragma unroll
  for (int j = 0; j < 4; ++j) out[obase + j] = acc[j];
}

// ---------------------------------------------------------------------------
extern "C" void kernel_launch(void* const* d_in, const int* in_sizes, int n_in,
                              void* d_out, int out_size, void* d_ws,
                              size_t ws_size, hipStream_t stream) {
  const float* qs = (const float*)d_in[0];
  const float* ks = (const float*)d_in[1];
  const float* Wq = (const float*)d_in[2];
  const float* Wk = (const float*)d_in[3];
  const float* wv = (const float*)d_in[4];
  float* out = (float*)d_out;

  float* Qp = (float*)d_ws;                          // [2048,128] f32 = 1 MB
  float* Kp = Qp + (size_t)MROWS * H_DIM;            // [2048,128] f32 = 1 MB

  dim3 g1(MROWS / 16, 2);
  proj_wmma_f32<<<g1, 256, 0, stream>>>(qs, ks, Wq, Wk, Qp, Kp);

  dim3 g2(LK_DIM / 64, LQ_DIM / 16, BATCH);
  additive_score<<<g2, 256, 0, stream>>>(Qp, Kp, wv, out);
}